// Decoder_46660524704357
// MI455X (gfx1250) — compile-verified
//
#include <hip/hip_runtime.h>
#include <hip/hip_bf16.h>

// Problem constants (match reference)
#define B_ 64
#define C_ 23
#define L_ 26000
#define D_ 64
#define E_ 512
#define K_ 3
// workspace layout in floats:
//   [0, 8192)          M2 padded (E=512 rows x 16 cols, cols 3..15 = 0)
//   [8192, 8208)       bk (3 used)
//   [8208, 8208+4416)  e  (1472 rows x 3)
#define WS_M2 0
#define WS_BK 8192
#define WS_E  8208

typedef __attribute__((ext_vector_type(2))) float v2f;
typedef __attribute__((ext_vector_type(4))) float v4f;
typedef __attribute__((ext_vector_type(8))) float v8f;
typedef __attribute__((ext_vector_type(4))) int   v4i;

// ---------------------------------------------------------------------------
// Kernel 1: fold eos_W @ Weᵀ into M2 (E x 16, zero padded), and bias bk.
// ---------------------------------------------------------------------------
__global__ void prep_kernel(const float* __restrict__ eos_W,   // (D,E) row-major
                            const float* __restrict__ eos_b,   // (D,)
                            const float* __restrict__ fc_W,    // (K, 2D) row-major
                            const float* __restrict__ fc_b,    // (K,)
                            float* __restrict__ ws) {
  const int e = threadIdx.x;            // 0..511
  float* M2 = ws + WS_M2;
  float s0 = 0.f, s1 = 0.f, s2 = 0.f;
  #pragma unroll 8
  for (int d = 0; d < D_; ++d) {
    const float w = eos_W[d * E_ + e];
    s0 += w * fc_W[0 * 2 * D_ + D_ + d];
    s1 += w * fc_W[1 * 2 * D_ + D_ + d];
    s2 += w * fc_W[2 * 2 * D_ + D_ + d];
  }
  M2[e * 16 + 0] = s0;
  M2[e * 16 + 1] = s1;
  M2[e * 16 + 2] = s2;
  #pragma unroll
  for (int k = 3; k < 16; ++k) M2[e * 16 + k] = 0.f;

  if (e == 0) {
    float* bk = ws + WS_BK;
    #pragma unroll
    for (int k = 0; k < K_; ++k) {
      float t = fc_b[k];
      for (int d = 0; d < D_; ++d) t += eos_b[d] * fc_W[k * 2 * D_ + D_ + d];
      bk[k] = t;
    }
    for (int k = K_; k < 16; ++k) bk[k] = 0.f;
  }
}

// ---------------------------------------------------------------------------
// Kernel 2: e = eos_emb(1472x512) @ M2(512x16) via V_WMMA_F32_16X16X4_F32.
// One wave (32 lanes) per 16-row tile; 1472/16 = 92 tiles. 128 WMMAs / wave.
// A-frag layout (ISA 7.12.2, 32-bit A 16x4): lane m(<16): K=0,1; lane m+16: K=2,3.
// B-frag: lanes 0-15 hold rows K=0(V0),K=1(V1); lanes 16-31 rows K=2,K=3.
// D-frag: VGPR r holds row r + 8*(lane>>4), col = lane&15.
// ---------------------------------------------------------------------------
__global__ void e_wmma_kernel(const float* __restrict__ eos_emb, // (1472, 512)
                              float* __restrict__ ws) {
  const float* M2 = ws + WS_M2;
  const float* bk = ws + WS_BK;
  float* eout = ws + WS_E;

  const int tile = blockIdx.x;        // 0..91
  const int lane = threadIdx.x;       // 0..31
  const int m = lane & 15;
  const int half = lane >> 4;

  const float* A = eos_emb + (size_t)(tile * 16 + m) * E_;

  v8f acc = {};
  for (int kk = 0; kk < E_; kk += 4) {
    const int kb = kk + half * 2;
    v2f a;
    a.x = A[kb];
    a.y = A[kb + 1];
    v2f b;
    b.x = M2[(kb + 0) * 16 + m];
    b.y = M2[(kb + 1) * 16 + m];
    acc = __builtin_amdgcn_wmma_f32_16x16x4_f32(
        /*neg_a=*/false, a, /*neg_b=*/false, b,
        /*c_mod=*/(short)0, acc, /*reuse_a=*/false, /*reuse_b=*/false);
  }

  // write only the 3 real columns
  if (m < K_) {
    const float bias = bk[m];
    #pragma unroll
    for (int r = 0; r < 8; ++r) {
      const int row = tile * 16 + r + 8 * half;
      eout[row * K_ + m] = acc[r] + bias;
    }
  }
}

// ---------------------------------------------------------------------------
// Kernel 3: the bandwidth kernel.
// out[b,c,l,k] = relu( dot(emb_table[bin_ids[c,l]], Wb[k]) + e[b,c,k] )
// One thread per 4 consecutive l (L % 4 == 0). Per b-iteration each thread
// emits 48 contiguous bytes as three aligned v4f NON-TEMPORAL stores
// (output is 459 MB write-once; bypass L2 rinse). emb_table rows are
// read-once -> non-temporal loads.
// ---------------------------------------------------------------------------
__global__ void out_kernel(const int* __restrict__ bin_ids,     // (C, L)
                           const float* __restrict__ emb_table, // (V, D)
                           const float* __restrict__ fc_W,      // (K, 2D)
                           const float* __restrict__ ws,
                           float* __restrict__ out) {
  __shared__ __align__(16) float sWb[K_ * D_];  // 192 floats
  __shared__ __align__(16) float sE[B_ * 4];    // e for this c, padded to 4/row

  const float* e_ws = ws + WS_E;
  const int c = blockIdx.y;
  const int tid = threadIdx.x;

  // Stage Wb (192 f) and e[:,c,:] (192 f -> padded 256 f) into LDS.
  for (int i = tid; i < K_ * D_ + B_ * K_; i += blockDim.x) {
    if (i < K_ * D_) {
      sWb[i] = fc_W[(i / D_) * 2 * D_ + (i % D_)];
    } else {
      const int j = i - K_ * D_;
      const int b = j / K_, k = j % K_;
      sE[b * 4 + k] = e_ws[(b * C_ + c) * K_ + k];
    }
  }
  __syncthreads();

  const int l0 = (blockIdx.x * blockDim.x + tid) * 4;  // group of 4 l's
  if (l0 >= L_) return;

  // 4 bin ids, one aligned 16B load
  const v4i bid = *(const v4i*)(bin_ids + c * L_ + l0);
  const v4f* row0 = (const v4f*)(emb_table + (size_t)bid.x * D_);
  const v4f* row1 = (const v4f*)(emb_table + (size_t)bid.y * D_);
  const v4f* row2 = (const v4f*)(emb_table + (size_t)bid.z * D_);
  const v4f* row3 = (const v4f*)(emb_table + (size_t)bid.w * D_);

  float a[4][K_];
  #pragma unroll
  for (int i = 0; i < 4; ++i)
    #pragma unroll
    for (int k = 0; k < K_; ++k) a[i][k] = 0.f;

  #pragma unroll
  for (int j = 0; j < D_ / 4; ++j) {
    const v4f w0 = *(const v4f*)&sWb[0 * D_ + 4 * j];
    const v4f w1 = *(const v4f*)&sWb[1 * D_ + 4 * j];
    const v4f w2 = *(const v4f*)&sWb[2 * D_ + 4 * j];
    const v4f r0 = __builtin_nontemporal_load(row0 + j);
    const v4f r1 = __builtin_nontemporal_load(row1 + j);
    const v4f r2 = __builtin_nontemporal_load(row2 + j);
    const v4f r3 = __builtin_nontemporal_load(row3 + j);
    a[0][0] += r0.x*w0.x + r0.y*w0.y + r0.z*w0.z + r0.w*w0.w;
    a[0][1] += r0.x*w1.x + r0.y*w1.y + r0.z*w1.z + r0.w*w1.w;
    a[0][2] += r0.x*w2.x + r0.y*w2.y + r0.z*w2.z + r0.w*w2.w;
    a[1][0] += r1.x*w0.x + r1.y*w0.y + r1.z*w0.z + r1.w*w0.w;
    a[1][1] += r1.x*w1.x + r1.y*w1.y + r1.z*w1.z + r1.w*w1.w;
    a[1][2] += r1.x*w2.x + r1.y*w2.y + r1.z*w2.z + r1.w*w2.w;
    a[2][0] += r2.x*w0.x + r2.y*w0.y + r2.z*w0.z + r2.w*w0.w;
    a[2][1] += r2.x*w1.x + r2.y*w1.y + r2.z*w1.z + r2.w*w1.w;
    a[2][2] += r2.x*w2.x + r2.y*w2.y + r2.z*w2.z + r2.w*w2.w;
    a[3][0] += r3.x*w0.x + r3.y*w0.y + r3.z*w0.z + r3.w*w0.w;
    a[3][1] += r3.x*w1.x + r3.y*w1.y + r3.z*w1.z + r3.w*w1.w;
    a[3][2] += r3.x*w2.x + r3.y*w2.y + r3.z*w2.z + r3.w*w2.w;
  }

  const size_t bstride = (size_t)C_ * L_ * K_;  // element stride between b's
  float* op = out + ((size_t)c * L_ + l0) * K_; // 48B-aligned (l0 % 4 == 0)

  #pragma unroll 2
  for (int b = 0; b < B_; ++b) {
    const v4f e4 = *(const v4f*)&sE[b * 4];
    v4f* p = (v4f*)(op + (size_t)b * bstride);
    v4f o0, o1, o2;  // 12 outputs: [l0+i][k], i=0..3, k=0..2
    o0.x = fmaxf(a[0][0] + e4.x, 0.f);
    o0.y = fmaxf(a[0][1] + e4.y, 0.f);
    o0.z = fmaxf(a[0][2] + e4.z, 0.f);
    o0.w = fmaxf(a[1][0] + e4.x, 0.f);
    o1.x = fmaxf(a[1][1] + e4.y, 0.f);
    o1.y = fmaxf(a[1][2] + e4.z, 0.f);
    o1.z = fmaxf(a[2][0] + e4.x, 0.f);
    o1.w = fmaxf(a[2][1] + e4.y, 0.f);
    o2.x = fmaxf(a[2][2] + e4.z, 0.f);
    o2.y = fmaxf(a[3][0] + e4.x, 0.f);
    o2.z = fmaxf(a[3][1] + e4.y, 0.f);
    o2.w = fmaxf(a[3][2] + e4.z, 0.f);
    __builtin_nontemporal_store(o0, p + 0);
    __builtin_nontemporal_store(o1, p + 1);
    __builtin_nontemporal_store(o2, p + 2);
  }
}

// ---------------------------------------------------------------------------
extern "C" void kernel_launch(void* const* d_in, const int* in_sizes, int n_in,
                              void* d_out, int out_size, void* d_ws, size_t ws_size,
                              hipStream_t stream) {
  const float* eos_emb   = (const float*)d_in[0];  // (B, C, E)
  const int*   bin_ids   = (const int*)  d_in[1];  // (C, L)
  const float* emb_table = (const float*)d_in[2];  // (V, D)
  const float* eos_W     = (const float*)d_in[3];  // (D, E)
  const float* eos_b     = (const float*)d_in[4];  // (D,)
  const float* fc_W      = (const float*)d_in[5];  // (K, 2D)
  const float* fc_b      = (const float*)d_in[6];  // (K,)
  float* ws  = (float*)d_ws;
  float* out = (float*)d_out;

  hipLaunchKernelGGL(prep_kernel, dim3(1), dim3(E_), 0, stream,
                     eos_W, eos_b, fc_W, fc_b, ws);

  hipLaunchKernelGGL(e_wmma_kernel, dim3((B_ * C_) / 16), dim3(32), 0, stream,
                     eos_emb, ws);

  const int lgroups = L_ / 4;                       // 6500
  dim3 grid((lgroups + 255) / 256, C_);             // (26, 23)
  hipLaunchKernelGGL(out_kernel, grid, dim3(256), 0, stream,
                     bin_ids, emb_table, fc_W, ws, out);
}